// OHEM_loss_8581344657452
// MI455X (gfx1250) — compile-verified
//
#include <hip/hip_runtime.h>

typedef float v2f __attribute__((ext_vector_type(2)));
typedef float v8f __attribute__((ext_vector_type(8)));

#define THREADS 256
#define MAXBLOCKS 2048

__device__ __forceinline__ float smooth_l1_term(float d) {
    float ad = fabsf(d);
    return (ad < 1.0f) ? 0.5f * d * d : ad - 0.5f;
}

__global__ void ohem_init_kernel(double* acc) {
    if (threadIdx.x == 0) { acc[0] = 0.0; acc[1] = 0.0; }
}

__global__ __launch_bounds__(THREADS) void ohem_main_kernel(
    const float* __restrict__ loc_preds,
    const float* __restrict__ loc_targets,
    const int*   __restrict__ cls_targets,
    double* __restrict__ acc,
    int n_anchors)
{
    __shared__ float sLoss[THREADS];
    __shared__ float sCnt[THREADS];

    const int tid    = threadIdx.x;
    const int stride = gridDim.x * blockDim.x;

    float loss = 0.0f;
    float cnt  = 0.0f;

    for (int i = blockIdx.x * blockDim.x + tid; i < n_anchors; i += stride) {
        // prefetch next grid-stride iteration (gfx1250 global_prefetch_b8)
        int ni = i + stride;
        if (ni < n_anchors) {
            __builtin_prefetch(loc_preds   + (size_t)ni * 8, 0, 1);
            __builtin_prefetch(loc_targets + (size_t)ni * 8, 0, 1);
            __builtin_prefetch(cls_targets + ni, 0, 1);
        }

        int   t    = cls_targets[i];
        float posf = (t > 0) ? 1.0f : 0.0f;

        const float4* p = (const float4*)(loc_preds   + (size_t)i * 8);
        const float4* q = (const float4*)(loc_targets + (size_t)i * 8);
        float4 p0 = p[0], p1 = p[1];
        float4 q0 = q[0], q1 = q[1];

        float s = 0.0f;
        s += smooth_l1_term(p0.x - q0.x);
        s += smooth_l1_term(p0.y - q0.y);
        s += smooth_l1_term(p0.z - q0.z);
        s += smooth_l1_term(p0.w - q0.w);
        s += smooth_l1_term(p1.x - q1.x);
        s += smooth_l1_term(p1.y - q1.y);
        s += smooth_l1_term(p1.z - q1.z);
        s += smooth_l1_term(p1.w - q1.w);

        loss += s * posf;
        cnt  += posf;
    }

    sLoss[tid] = loss;
    sCnt[tid]  = cnt;
    __syncthreads();

    // Block reduction of 256 partials via chained V_WMMA_F32_16X16X4_F32:
    // A = 16x4 tile of partials (2 VGPRs/lane), B = all-ones 4x16,
    // C accumulates across 4 WMMAs -> D[m][n] = sum of 16 partials per row m.
    if (tid < 32) {                      // wave 0: EXEC all ones inside
        v2f ones; ones.x = 1.0f; ones.y = 1.0f;
        v8f cl = {0.f,0.f,0.f,0.f,0.f,0.f,0.f,0.f};
        v8f cc = {0.f,0.f,0.f,0.f,0.f,0.f,0.f,0.f};
#pragma unroll
        for (int blk = 0; blk < 4; ++blk) {
            int base = blk * 64;
            v2f a; a.x = sLoss[base + 2*tid]; a.y = sLoss[base + 2*tid + 1];
            cl = __builtin_amdgcn_wmma_f32_16x16x4_f32(
                    false, a, false, ones, (short)0, cl, false, false);
            v2f b; b.x = sCnt[base + 2*tid];  b.y = sCnt[base + 2*tid + 1];
            cc = __builtin_amdgcn_wmma_f32_16x16x4_f32(
                    false, b, false, ones, (short)0, cc, false, false);
        }
        // lanes 0-15 hold rows 0-7 (8 VGPRs), lanes 16-31 hold rows 8-15
        float tl = cl[0]+cl[1]+cl[2]+cl[3]+cl[4]+cl[5]+cl[6]+cl[7];
        float tc = cc[0]+cc[1]+cc[2]+cc[3]+cc[4]+cc[5]+cc[6]+cc[7];
        tl += __shfl_xor(tl, 16, 32);
        tc += __shfl_xor(tc, 16, 32);
        if (tid == 0) {
            atomicAdd(&acc[0], (double)tl);   // global_atomic_add_f64
            atomicAdd(&acc[1], (double)tc);
        }
    }
}

__global__ void ohem_final_kernel(const double* __restrict__ acc,
                                  float* __restrict__ out) {
    if (threadIdx.x == 0) {
        double N = acc[1];
        out[0] = (float)(0.2 * acc[0] / N);
    }
}

extern "C" void kernel_launch(void* const* d_in, const int* in_sizes, int n_in,
                              void* d_out, int out_size, void* d_ws, size_t ws_size,
                              hipStream_t stream) {
    const float* loc_preds   = (const float*)d_in[0];
    const float* loc_targets = (const float*)d_in[1];
    // d_in[2] = cls_preds: not needed — ce == 0 exactly when NUM_CLASSES == 1
    const int*   cls_targets = (const int*)d_in[3];

    int n_anchors = in_sizes[3];               // B * A
    double* acc = (double*)d_ws;               // [0]=loc_loss sum, [1]=num_pos

    int blocks = (n_anchors + THREADS - 1) / THREADS;
    if (blocks > MAXBLOCKS) blocks = MAXBLOCKS;
    if (blocks < 1) blocks = 1;

    ohem_init_kernel<<<1, 32, 0, stream>>>(acc);
    ohem_main_kernel<<<blocks, THREADS, 0, stream>>>(
        loc_preds, loc_targets, cls_targets, acc, n_anchors);
    ohem_final_kernel<<<1, 32, 0, stream>>>(acc, (float*)d_out);
}